// OutputBlock_53833120088086
// MI455X (gfx1250) — compile-verified
//
#include <hip/hip_runtime.h>
#include <hip/hip_bf16.h>

typedef float v2f __attribute__((ext_vector_type(2)));
typedef float v8f __attribute__((ext_vector_type(8)));

#define EMBED     128
#define OUT_EMBED 256
#define NUM_RBF   6

// ---------------------------------------------------------------------------
// Zero the f32 accumulator (must run every launch; harness does not re-poison)
// ---------------------------------------------------------------------------
__global__ void zero_f32_kernel(float* __restrict__ p, long long n) {
  long long i = (long long)blockIdx.x * blockDim.x + threadIdx.x;
  if (i < n) p[i] = 0.0f;
}

// ---------------------------------------------------------------------------
// Stage 1: gated = messages * (rbf @ W_rbf); scatter-add into summed[idx_i]
// One wave32 per edge; each lane owns 4 consecutive columns (float4).
// Bandwidth-bound: 328 MB streaming read, atomics land in the 10MB L2-resident
// accumulator.
// ---------------------------------------------------------------------------
__global__ void gate_scatter_kernel(const float* __restrict__ msg,
                                    const float* __restrict__ rbf,
                                    const int*   __restrict__ idx,
                                    const float* __restrict__ Wrbf,   // [6][128]
                                    float*       __restrict__ acc,    // [P][128]
                                    int n_edges) {
  int t = blockIdx.x * blockDim.x + threadIdx.x;
  int e = t >> 5;
  if (e >= n_edges) return;
  int lane = t & 31;
  int c0 = lane << 2;  // 4 columns per lane -> 128 columns per wave

  float r[NUM_RBF];
#pragma unroll
  for (int j = 0; j < NUM_RBF; ++j) r[j] = rbf[(size_t)e * NUM_RBF + j];

  float4 w = make_float4(0.f, 0.f, 0.f, 0.f);
#pragma unroll
  for (int j = 0; j < NUM_RBF; ++j) {
    const float4 wr = *(const float4*)(Wrbf + j * EMBED + c0);  // 3KB, cache-hot
    w.x = fmaf(r[j], wr.x, w.x);
    w.y = fmaf(r[j], wr.y, w.y);
    w.z = fmaf(r[j], wr.z, w.z);
    w.w = fmaf(r[j], wr.w, w.w);
  }

  const float4 m4 = *(const float4*)(msg + (size_t)e * EMBED + c0);
  float* dst = acc + (size_t)idx[e] * EMBED + c0;
  atomicAdd(dst + 0, m4.x * w.x);
  atomicAdd(dst + 1, m4.y * w.y);
  atomicAdd(dst + 2, m4.z * w.z);
  atomicAdd(dst + 3, m4.w * w.w);
}

// ---------------------------------------------------------------------------
// Stage 2-4: fp32 GEMM via V_WMMA_F32_16X16X4_F32.
// One wave32 computes one 16x16 output tile; K-loop in steps of 4.
// A 16x4 layout: M = lane&15; VGPR pair holds K = {0,1} (lanes 0-15) or {2,3}
// (lanes 16-31). B 4x16 mirrored over N. C/D: VGPR j -> M = j + 8*(lane>>4),
// N = lane&15. Optional fused bias + swish epilogue.
// ---------------------------------------------------------------------------
template <bool SWISH>
__global__ void __launch_bounds__(256)
gemm_wmma_f32_kernel(const float* __restrict__ A,     // [M][K]
                     const float* __restrict__ B,     // [K][N]
                     const float* __restrict__ bias,  // [N] or nullptr
                     float*       __restrict__ C,     // [M][N]
                     int M, int N, int K, int totalTiles) {
  const int wave = blockIdx.x * (blockDim.x >> 5) + (threadIdx.x >> 5);
  if (wave >= totalTiles) return;  // wave-uniform: EXEC all-1s inside WMMA
  const int tilesN = N >> 4;
  const int tm = wave / tilesN;
  const int tn = wave % tilesN;

  const int lane = threadIdx.x & 31;
  const int half = lane >> 4;   // 0: K lo pair, 1: K hi pair
  const int lid  = lane & 15;

  const float* ap = A + (size_t)(tm * 16 + lid) * K + 2 * half;
  const float* bp = B + (size_t)(2 * half) * N + tn * 16 + lid;

  v8f acc = {};
  for (int k = 0; k < K; k += 4) {
    v2f a; a.x = ap[0]; a.y = ap[1];        // A[m][k+2h], A[m][k+2h+1]
    v2f b; b.x = bp[0]; b.y = bp[N];        // B[k+2h][n], B[k+2h+1][n]
    acc = __builtin_amdgcn_wmma_f32_16x16x4_f32(
        /*neg_a=*/false, a, /*neg_b=*/false, b,
        /*c_mod=*/(short)0, acc, /*reuse_a=*/false, /*reuse_b=*/false);
    ap += 4;
    bp += 4 * N;
  }

  const int n = tn * 16 + lid;
  const float bv = bias ? bias[n] : 0.0f;
#pragma unroll
  for (int j = 0; j < 8; ++j) {
    const int m = tm * 16 + j + 8 * half;
    float x = acc[j] + bv;
    if (SWISH) x = x * (1.0f / (1.0f + __expf(-x)));  // swish = x*sigmoid(x)
    C[(size_t)m * N + n] = x;
  }
}

// ---------------------------------------------------------------------------
// Stage 5: out = H @ W_final  (256 -> 1). Wave per row + shuffle reduction.
// ---------------------------------------------------------------------------
__global__ void matvec_final_kernel(const float* __restrict__ H,
                                    const float* __restrict__ Wf,
                                    float* __restrict__ out, int M, int N) {
  int t = blockIdx.x * blockDim.x + threadIdx.x;
  int row = t >> 5;
  if (row >= M) return;
  int lane = t & 31;
  const float* h = H + (size_t)row * N;
  float s = 0.0f;
  for (int n = lane; n < N; n += 32) s = fmaf(h[n], Wf[n], s);
#pragma unroll
  for (int off = 16; off > 0; off >>= 1) s += __shfl_down(s, off, 32);
  if (lane == 0) out[row] = s;
}

// ---------------------------------------------------------------------------
extern "C" void kernel_launch(void* const* d_in, const int* in_sizes, int n_in,
                              void* d_out, int out_size, void* d_ws, size_t ws_size,
                              hipStream_t stream) {
  const float* msg  = (const float*)d_in[0];  // [E,128]
  const float* rbf  = (const float*)d_in[1];  // [E,6]
  const int*   idx  = (const int*)  d_in[2];  // [E]
  //            d_in[3] = n_particles scalar (device); P derived from out_size
  const float* Wrbf = (const float*)d_in[4];  // [6,128]
  const float* Wup  = (const float*)d_in[5];  // [128,256]
  const float* Wd   = (const float*)d_in[6];  // [3,256,256]
  const float* bd   = (const float*)d_in[7];  // [3,256]
  const float* Wf   = (const float*)d_in[8];  // [256,1]
  float* out = (float*)d_out;

  const int n_edges = in_sizes[0] / EMBED;
  const int P = out_size;  // NUM_TARGETS == 1

  // Workspace layout (all fp32): summed [P,128] | H0 [P,256] | H1 [P,256]
  float* summed = (float*)d_ws;
  size_t szSum  = ((size_t)P * EMBED * sizeof(float) + 255) & ~(size_t)255;
  float* H0 = (float*)((char*)d_ws + szSum);
  float* H1 = H0 + (size_t)P * OUT_EMBED;

  // 1) zero accumulator
  {
    long long n = (long long)P * EMBED;
    int blocks = (int)((n + 255) / 256);
    zero_f32_kernel<<<blocks, 256, 0, stream>>>(summed, n);
  }

  // 2) gate + scatter-add (wave per edge)
  {
    long long threads = (long long)n_edges * 32;
    int blocks = (int)((threads + 255) / 256);
    gate_scatter_kernel<<<blocks, 256, 0, stream>>>(msg, rbf, idx, Wrbf, summed,
                                                    n_edges);
  }

  // 3) H0 = summed @ W_up   [P,128] x [128,256]
  {
    int tiles = ((P + 15) / 16) * (OUT_EMBED / 16);
    int blocks = (tiles + 7) / 8;  // 8 waves per 256-thread block
    gemm_wmma_f32_kernel<false><<<blocks, 256, 0, stream>>>(
        summed, Wup, nullptr, H0, P, OUT_EMBED, EMBED, tiles);
  }

  // 4) three swish dense layers: H = swish(H @ Wd[i] + bd[i])
  {
    int tiles = ((P + 15) / 16) * (OUT_EMBED / 16);
    int blocks = (tiles + 7) / 8;
    const float* src = H0;
    float* dst = H1;
    for (int i = 0; i < 3; ++i) {
      gemm_wmma_f32_kernel<true><<<blocks, 256, 0, stream>>>(
          src, Wd + (size_t)i * OUT_EMBED * OUT_EMBED, bd + (size_t)i * OUT_EMBED,
          dst, P, OUT_EMBED, OUT_EMBED, tiles);
      const float* tmp = src; src = dst; dst = (float*)tmp;
    }
    // after 3 swaps, result lives in H1 (H0->H1->H0->H1? trace: i0: H0->H1,
    // i1: H1->H0, i2: H0->H1) => final in H1
  }

  // 5) out = H1 @ W_final   [P,256] x [256,1]
  {
    long long threads = (long long)P * 32;
    int blocks = (int)((threads + 255) / 256);
    matvec_final_kernel<<<blocks, 256, 0, stream>>>(H1, Wf, out, P, OUT_EMBED);
  }
}